// RPNPostProcessor_12532714570350
// MI455X (gfx1250) — compile-verified
//
#include <hip/hip_runtime.h>
#include <hip/hip_bf16.h>
#include <math.h>

typedef float v2f __attribute__((ext_vector_type(2)));
typedef float v8f __attribute__((ext_vector_type(8)));

#if defined(__has_builtin)
#if __has_builtin(__builtin_amdgcn_wmma_f32_16x16x4_f32)
#define HAVE_WMMA_F32_16X16X4 1
#endif
#endif

#define NB 4
#define NA 15
#define NH 200
#define NW 200
#define MTOT (NA * NH * NW) /* 600000 */
#define KTOP 1000
#define KPAD 1024
#define POST_TOP 300
#define NMS_T 0.7f
#define MIN_SIZE 4.0f
#define BBOX_CLIP 4.135166556742356f     /* log(1000/16) */
#define RAD2DEG 57.29577951308232f
#define DEG2RAD 0.017453292519943295f

// ---------- helpers ----------
__device__ __forceinline__ unsigned tokey(float f) {
  unsigned u = __float_as_uint(f);
  return (u & 0x80000000u) ? ~u : (u | 0x80000000u);
}
__device__ __forceinline__ float fromkey(unsigned k) {
  unsigned u = (k & 0x80000000u) ? (k & 0x7FFFFFFFu) : ~k;
  return __uint_as_float(u);
}

// Sutherland-Hodgman: clip quad A (corners at rowA[4..11]) by quad B edges.
// Matches reference math: inside = side >= 0, denom clamp 1e-12 (positive).
__device__ float quad_inter(const float* rowA, const float* rowB) {
  float px[8], py[8], qx[8], qy[8];
#pragma unroll
  for (int k = 0; k < 4; ++k) { px[k] = rowA[4 + 2 * k]; py[k] = rowA[5 + 2 * k]; }
#pragma unroll
  for (int k = 4; k < 8; ++k) { px[k] = 0.f; py[k] = 0.f; }
  int nv = 4;
#pragma unroll
  for (int e = 0; e < 4; ++e) {
    float p1x = rowB[4 + 2 * e], p1y = rowB[5 + 2 * e];
    int e2 = (e + 1) & 3;
    float p2x = rowB[4 + 2 * e2], p2y = rowB[5 + 2 * e2];
    float dx = p2x - p1x, dy = p2y - p1y;
    int m = 0;
    for (int k = 0; k < 8; ++k) {
      if (k >= nv) break;
      int kn = (k + 1 == nv) ? 0 : k + 1;
      float sc = dx * (py[k] - p1y) - dy * (px[k] - p1x);
      float sn = dx * (py[kn] - p1y) - dy * (px[kn] - p1x);
      bool ic = sc >= 0.f, in2 = sn >= 0.f;
      if (ic && m < 8) { qx[m] = px[k]; qy[m] = py[k]; ++m; }
      if ((ic != in2) && m < 8) {
        float den = sc - sn;
        if (fabsf(den) < 1e-12f) den = 1e-12f;
        float tt = sc / den;
        qx[m] = px[k] + tt * (px[kn] - px[k]);
        qy[m] = py[k] + tt * (py[kn] - py[k]);
        ++m;
      }
    }
    for (int k = 0; k < 8; ++k) {
      px[k] = (k < m) ? qx[k] : 0.f;
      py[k] = (k < m) ? qy[k] : 0.f;
    }
    nv = m;
  }
  float s = 0.f;
  for (int k = 0; k < 8; ++k) {
    if (k < nv) {
      int kn = (k + 1 == nv) ? 0 : k + 1;
      s += px[k] * py[kn] - px[kn] * py[k];
    }
  }
  return 0.5f * fabsf(s);
}

// ---------- kernels ----------
__global__ void init_kernel(unsigned* hist, unsigned* prefix, int* kr,
                            unsigned* cntG, unsigned* cntE, unsigned* validw) {
  int t = threadIdx.x;
  if (t < NB * 256) hist[t] = 0u;
  if (t < NB) { prefix[t] = 0u; kr[t] = KTOP; cntG[t] = 0u; cntE[t] = 0u; }
  if (t < NB * 32) validw[t] = 0u;
}

__global__ void hist_kernel(const float* __restrict__ obj, unsigned* hist,
                            const unsigned* __restrict__ prefix, int byte) {
  __shared__ unsigned lh[256];
  int n = blockIdx.y;
  lh[threadIdx.x] = 0u;
  __syncthreads();
  const float* base = obj + (size_t)n * MTOT;
  unsigned pfx = prefix[n];
  int start = blockIdx.x * (256 * 8) + threadIdx.x;
  for (int e = 0; e < 8; ++e) {
    int idx = start + e * 256;
    if (idx >= MTOT) break;
    unsigned key = tokey(base[idx]);
    bool match = true;
    if (byte < 3) {
      int sh = (byte + 1) * 8;
      match = (key >> sh) == (pfx >> sh);
    }
    if (match) atomicAdd(&lh[(key >> (byte * 8)) & 255u], 1u);
  }
  __syncthreads();
  if (lh[threadIdx.x]) atomicAdd(&hist[n * 256 + threadIdx.x], lh[threadIdx.x]);
}

__global__ void scan_kernel(unsigned* hist, unsigned* prefix, int* kr, int byte) {
  int n = blockIdx.x;
  if (threadIdx.x == 0) {
    unsigned need = (unsigned)kr[n];
    unsigned c = 0;
    for (int bin = 255; bin >= 0; --bin) {
      unsigned h = hist[n * 256 + bin];
      if (c + h >= need) {
        prefix[n] |= ((unsigned)bin) << (byte * 8);
        kr[n] = (int)(need - c);
        break;
      }
      c += h;
    }
  }
  __syncthreads();
  hist[n * 256 + threadIdx.x] = 0u;
}

__global__ void gather_kernel(const float* __restrict__ obj, const unsigned* __restrict__ prefix,
                              const int* __restrict__ kr, unsigned* cntG, unsigned* cntE,
                              unsigned* candkey, unsigned* candidx) {
  int n = blockIdx.y;
  unsigned T = prefix[n];
  int Eneed = kr[n];
  const float* base = obj + (size_t)n * MTOT;
  int start = blockIdx.x * (256 * 8) + threadIdx.x;
  for (int e = 0; e < 8; ++e) {
    int idx = start + e * 256;
    if (idx >= MTOT) break;
    unsigned key = tokey(base[idx]);
    if (key > T) {
      unsigned p = atomicAdd(&cntG[n], 1u);
      if (p < KTOP) {
        unsigned a = (unsigned)idx / (NH * NW);
        unsigned hw = (unsigned)idx % (NH * NW);
        candkey[n * KPAD + p] = key;
        candidx[n * KPAD + p] = hw * NA + a;
      }
    } else if (key == T) {
      unsigned p = atomicAdd(&cntE[n], 1u);
      if ((int)p < Eneed) {
        unsigned slot = (unsigned)(KTOP - 1) - p;
        unsigned a = (unsigned)idx / (NH * NW);
        unsigned hw = (unsigned)idx % (NH * NW);
        candkey[n * KPAD + slot] = key;
        candidx[n * KPAD + slot] = hw * NA + a;
      }
    }
  }
}

__global__ __launch_bounds__(1024) void sort_kernel(const unsigned* __restrict__ candkey,
                                                    const unsigned* __restrict__ candidx,
                                                    unsigned* selkey, unsigned* selidx) {
  __shared__ unsigned sk[KPAD];
  __shared__ unsigned si[KPAD];
  int n = blockIdx.x, t = threadIdx.x;
  if (t < KTOP) { sk[t] = candkey[n * KPAD + t]; si[t] = candidx[n * KPAD + t]; }
  else { sk[t] = 0u; si[t] = 0xFFFFFFFFu; }
  __syncthreads();
  for (int k = 2; k <= KPAD; k <<= 1) {
    for (int j = k >> 1; j > 0; j >>= 1) {
      int ixj = t ^ j;
      if (ixj > t) {
        unsigned ka = sk[t], kb = sk[ixj], ia = si[t], ib = si[ixj];
        bool aFirst = (ka > kb) || (ka == kb && ia < ib);  // descending, tie: low idx first
        bool desc = (t & k) == 0;
        if (desc ? !aFirst : aFirst) { sk[t] = kb; sk[ixj] = ka; si[t] = ib; si[ixj] = ia; }
      }
      __syncthreads();
    }
  }
  selkey[n * KPAD + t] = sk[t];
  selidx[n * KPAD + t] = si[t];
}

__global__ void decode_kernel(const float* __restrict__ breg, const float* __restrict__ anchors,
                              const unsigned* __restrict__ selidx, const unsigned* __restrict__ selkey,
                              float* props, float* scores, float* aux, unsigned* validw) {
  int g = blockIdx.x * blockDim.x + threadIdx.x;
  if (g >= NB * KPAD) return;
  int n = g >> 10, r = g & (KPAD - 1);
  float p0 = 0.f, p1 = 0.f, p2 = 0.f, p3 = 0.f, p4 = 0.f, score = 0.f;
  bool valid = false;
  if (r < KTOP) {
    unsigned f = selidx[n * KPAD + r];
    float val = fromkey(selkey[n * KPAD + r]);
    score = 1.f / (1.f + expf(-val));
    int a = (int)(f % NA);
    int hw = (int)(f / NA);
    int hh = hw / NW, ww = hw % NW;
    const float* anc = anchors + ((size_t)n * MTOT + f) * 5;
    float xa = anc[0], ya = anc[1], wa = anc[2], ha = anc[3], ta = anc[4];
    size_t cb = (((size_t)n * NA * 5 + (size_t)a * 5) * NH + hh) * NW + ww;
    size_t chan = (size_t)NH * NW;
    float dxv = breg[cb + 0 * chan];
    float dyv = breg[cb + 1 * chan];
    float dwv = breg[cb + 2 * chan];
    float dhv = breg[cb + 3 * chan];
    float dtv = breg[cb + 4 * chan];
    dwv = fminf(dwv, BBOX_CLIP);
    dhv = fminf(dhv, BBOX_CLIP);
    p0 = dxv * wa + xa;
    p1 = dyv * ha + ya;
    p2 = expf(dwv) * wa;
    p3 = expf(dhv) * ha;
    p4 = ta + dtv * RAD2DEG;
    valid = (p2 >= MIN_SIZE) && (p3 >= MIN_SIZE);
  }
  float* pp = props + (size_t)(n * KPAD + r) * 5;
  pp[0] = p0; pp[1] = p1; pp[2] = p2; pp[3] = p3; pp[4] = p4;
  scores[n * KPAD + r] = score;
  float* ax = aux + (size_t)(n * KPAD + r) * 13;
  float ar = p4 * DEG2RAD;
  float cs = cosf(ar), sn = sinf(ar);
  ax[0] = p0;
  ax[1] = p1;
  ax[2] = p0 * p0 + p1 * p1;
  ax[3] = 0.5f * sqrtf(p2 * p2 + p3 * p3);
  float hx[4] = {0.5f * p2, -0.5f * p2, -0.5f * p2, 0.5f * p2};
  float hy[4] = {0.5f * p3, 0.5f * p3, -0.5f * p3, -0.5f * p3};
#pragma unroll
  for (int k = 0; k < 4; ++k) {
    ax[4 + 2 * k] = p0 + cs * hx[k] - sn * hy[k];
    ax[5 + 2 * k] = p1 + sn * hx[k] + cs * hy[k];
  }
  ax[12] = p2 * p3;
  if (r < KTOP && valid) atomicOr(&validw[n * 32 + (r >> 5)], 1u << (r & 31));
}

// Block = 16 i-rows x 32 j-cols. Wave 0 runs the WMMA distance prefilter
// (center dot products via V_WMMA_F32_16X16X4_F32), then 512 threads clip
// only candidate pairs and ballot a 32-bit suppression word per row.
__global__ __launch_bounds__(512) void pairs_kernel(const float* __restrict__ aux, unsigned* sup) {
  __shared__ float I[16][13];
  __shared__ float J[32][13];
  __shared__ unsigned cand[16];
  int n = blockIdx.z, ti = blockIdx.x, wj = blockIdx.y;
  int t = threadIdx.x;
  for (int s = t; s < 48 * 13; s += 512) {
    int rec = s / 13, c = s % 13;
    if (rec < 16) I[rec][c] = aux[((size_t)n * KPAD + ti * 16 + rec) * 13 + c];
    else J[rec - 16][c] = aux[((size_t)n * KPAD + wj * 32 + (rec - 16)) * 13 + c];
  }
#if !HAVE_WMMA_F32_16X16X4
  if (t < 16) cand[t] = 0xFFFFFFFFu;  // no prefilter: all pairs candidates
#endif
  __syncthreads();
#if HAVE_WMMA_F32_16X16X4
  if (t < 32) {  // wave 0 only, EXEC all ones
    int lane = t;
    int rl = lane & 15;
    bool lo = lane < 16;
    // A 16x4 f32: lanes0-15 hold K=0 (v0) / K=1 (v1); lanes16-31 K=2/3 -> zero pad
    v2f a;
    a.x = lo ? I[rl][0] : 0.f;
    a.y = lo ? I[rl][1] : 0.f;
    unsigned acc = 0u;  // lane r (r<16) accumulates cand word for row r
#pragma unroll
    for (int h2 = 0; h2 < 2; ++h2) {
      int cb = h2 * 16;
      // B 4x16 f32: lanes0-15 = cols, rows K=0/1 across v0/v1; lanes16-31 K=2/3 -> zero pad
      v2f b;
      b.x = lo ? J[cb + rl][0] : 0.f;
      b.y = lo ? J[cb + rl][1] : 0.f;
      v8f c8 = {0.f, 0.f, 0.f, 0.f, 0.f, 0.f, 0.f, 0.f};
      v8f d = __builtin_amdgcn_wmma_f32_16x16x4_f32(false, a, false, b, (short)0, c8,
                                                    false, false);
#pragma unroll
      for (int rr = 0; rr < 8; ++rr) {
        int irow = rr + (lo ? 0 : 8);  // C/D layout: vgpr rr = row rr (lanes0-15) / rr+8
        float dot = d[rr];
        float ni = I[irow][2], nj = J[cb + rl][2];
        float dist2 = ni + nj - 2.f * dot;
        float rs = I[irow][3] + J[cb + rl][3];
        // magnitude-scaled slack: covers cancellation error in ni+nj-2*dot
        bool cnd = dist2 <= rs * rs + 1e-5f * (ni + nj) + 1.0f;
        unsigned m = (unsigned)__ballot(cnd);  // wave-uniform (SGPR)
        acc |= (lane == rr) ? ((m & 0xFFFFu) << cb) : 0u;
        acc |= (lane == rr + 8) ? ((m >> 16) << cb) : 0u;
      }
    }
    if (lane < 16) cand[lane] = acc;
  }
#endif
  __syncthreads();
  int r = t >> 5, c = t & 31;
  int i = ti * 16 + r, j = wj * 32 + c;
  bool cnd = (cand[r] >> c) & 1u;
  bool active = cnd && (j > i) && (i < KTOP) && (j < KTOP);
  bool bit = false;
  if (active) {
    float inter = quad_inter(&I[r][0], &J[c][0]);
    float uni = I[r][12] + J[c][12] - inter;
    float iou = inter / fmaxf(uni, 1e-8f);
    bit = iou > NMS_T;
  }
  unsigned m = (unsigned)__ballot(bit);
  if (c == 0) sup[((size_t)n * KPAD + i) * 32 + wj] = m;
}

__global__ void greedy_kernel(const unsigned* __restrict__ validw,
                              const unsigned* __restrict__ sup, unsigned* keepout) {
  int n = blockIdx.x;
  int lane = threadIdx.x;  // 32 threads = 1 wave; lane owns keep word `lane`
  unsigned keepm = validw[n * 32 + lane];
  for (int i = 0; i < KTOP; ++i) {
    unsigned kw = (unsigned)__shfl((int)keepm, i >> 5, 32);
    if ((kw >> (i & 31)) & 1u) keepm &= ~sup[((size_t)n * KPAD + i) * 32 + lane];
  }
  int cnt = __popc(keepm);
  int pre = 0;
  for (int l = 0; l < 32; ++l) {
    int c = __shfl(cnt, l, 32);
    if (l < lane) pre += c;
  }
  int allowed = POST_TOP - pre;
  if (allowed <= 0) keepm = 0u;
  else {
    while (__popc(keepm) > allowed) keepm &= ~(1u << (31 - __clz((int)keepm)));
  }
  keepout[n * 32 + lane] = keepm;
}

__global__ void writeout_kernel(const float* __restrict__ props, const float* __restrict__ scores,
                                const unsigned* __restrict__ keep, float* out) {
  int g = blockIdx.x * blockDim.x + threadIdx.x;
  if (g >= NB * KTOP) return;
  int n = g / KTOP, i = g % KTOP;
  float f = ((keep[n * 32 + (i >> 5)] >> (i & 31)) & 1u) ? 1.f : 0.f;
  const float* p = props + (size_t)(n * KPAD + i) * 5;
  float* o = out + (size_t)(n * KTOP + i) * 6;
  o[0] = p[0] * f;
  o[1] = p[1] * f;
  o[2] = p[2] * f;
  o[3] = p[3] * f;
  o[4] = p[4] * f;
  o[5] = scores[n * KPAD + i] * f;
}

extern "C" void kernel_launch(void* const* d_in, const int* in_sizes, int n_in,
                              void* d_out, int out_size, void* d_ws, size_t ws_size,
                              hipStream_t stream) {
  const float* obj = (const float*)d_in[0];
  const float* breg = (const float*)d_in[1];
  const float* anchors = (const float*)d_in[2];
  float* out = (float*)d_out;
  char* ws = (char*)d_ws;
  size_t off = 0;
  auto take = [&](size_t bytes) {
    size_t r = off;
    off += (bytes + 255) & ~(size_t)255;
    return r;
  };
  unsigned* hist = (unsigned*)(ws + take(NB * 256 * 4));
  unsigned* prefix = (unsigned*)(ws + take(64));
  int* kr = (int*)(ws + take(64));
  unsigned* cntG = (unsigned*)(ws + take(64));
  unsigned* cntE = (unsigned*)(ws + take(64));
  unsigned* candkey = (unsigned*)(ws + take((size_t)NB * KPAD * 4));
  unsigned* candidx = (unsigned*)(ws + take((size_t)NB * KPAD * 4));
  unsigned* selkey = (unsigned*)(ws + take((size_t)NB * KPAD * 4));
  unsigned* selidx = (unsigned*)(ws + take((size_t)NB * KPAD * 4));
  float* props = (float*)(ws + take((size_t)NB * KPAD * 5 * 4));
  float* scores = (float*)(ws + take((size_t)NB * KPAD * 4));
  float* aux = (float*)(ws + take((size_t)NB * KPAD * 13 * 4));
  unsigned* validw = (unsigned*)(ws + take(NB * 32 * 4));
  unsigned* keep = (unsigned*)(ws + take(NB * 32 * 4));
  unsigned* sup = (unsigned*)(ws + take((size_t)NB * KPAD * 32 * 4));
  (void)in_sizes; (void)n_in; (void)out_size; (void)ws_size;

  init_kernel<<<1, 1024, 0, stream>>>(hist, prefix, kr, cntG, cntE, validw);
  int blocksPer = (MTOT + 256 * 8 - 1) / (256 * 8);
  for (int byte = 3; byte >= 0; --byte) {
    hist_kernel<<<dim3(blocksPer, NB), 256, 0, stream>>>(obj, hist, prefix, byte);
    scan_kernel<<<NB, 256, 0, stream>>>(hist, prefix, kr, byte);
  }
  gather_kernel<<<dim3(blocksPer, NB), 256, 0, stream>>>(obj, prefix, kr, cntG, cntE,
                                                         candkey, candidx);
  sort_kernel<<<NB, 1024, 0, stream>>>(candkey, candidx, selkey, selidx);
  decode_kernel<<<(NB * KPAD + 255) / 256, 256, 0, stream>>>(breg, anchors, selidx, selkey,
                                                             props, scores, aux, validw);
  pairs_kernel<<<dim3(KPAD / 16, KPAD / 32, NB), 512, 0, stream>>>(aux, sup);
  greedy_kernel<<<NB, 32, 0, stream>>>(validw, sup, keep);
  writeout_kernel<<<(NB * KTOP + 255) / 256, 256, 0, stream>>>(props, scores, keep, out);
}